// GateNetwork_68659347194377
// MI455X (gfx1250) — compile-verified
//
#include <hip/hip_runtime.h>
#include <hip/hip_bf16.h>

#define DDIM 4096
#define HALF_DIM 2048

typedef __attribute__((ext_vector_type(2))) float v2f;
typedef __attribute__((ext_vector_type(8))) float v8f;

// ---------------------------------------------------------------------------
// Kernel 1: gating. Computes 8 gate scores (4 rgb, 4 ir), picks top-1 of each
// pair, softmax over the two winners per branch, writes per-slot expert index
// and probability into workspace header, and the two argmax outputs.
// ws layout: int ws_i[0..3] = expert idx per slot (slot 0,1 = rgb; 2,3 = ir)
//            float ws_f[4..7] = prob per slot
// ---------------------------------------------------------------------------
__global__ __launch_bounds__(256) void gate_kernel(
    const float* __restrict__ rgb, const float* __restrict__ ir,
    const float* __restrict__ Wg_rgb, const float* __restrict__ bg_rgb,
    const float* __restrict__ Wg_ir,  const float* __restrict__ bg_ir,
    float* __restrict__ out, int* __restrict__ ws_i, float* __restrict__ ws_f)
{
    __shared__ float red[8][256];
    float part[8];
#pragma unroll
    for (int g = 0; g < 8; ++g) part[g] = 0.0f;

    for (int d = threadIdx.x; d < DDIM; d += 256) {
        float xv = (d < HALF_DIM) ? rgb[d] : ir[d - HALF_DIM];
#pragma unroll
        for (int g = 0; g < 4; ++g) {
            part[g]     = fmaf(xv, Wg_rgb[g * DDIM + d], part[g]);
            part[4 + g] = fmaf(xv, Wg_ir[g * DDIM + d],  part[4 + g]);
        }
    }
#pragma unroll
    for (int g = 0; g < 8; ++g) red[g][threadIdx.x] = part[g];
    __syncthreads();

    for (int stride = 128; stride > 0; stride >>= 1) {
        if (threadIdx.x < (unsigned)stride) {
#pragma unroll
            for (int g = 0; g < 8; ++g)
                red[g][threadIdx.x] += red[g][threadIdx.x + stride];
        }
        __syncthreads();
    }

    if (threadIdx.x == 0) {
        float s[8];
#pragma unroll
        for (int g = 0; g < 4; ++g) s[g]     = fmaxf(red[g][0]     + bg_rgb[g], 0.0f);
#pragma unroll
        for (int g = 0; g < 4; ++g) s[4 + g] = fmaxf(red[4 + g][0] + bg_ir[g],  0.0f);

        for (int br = 0; br < 2; ++br) {
            const float* sb = s + br * 4;
            // jnp.argmax returns first max index -> strict '>' for index 1
            int i1 = (sb[1] > sb[0]) ? 1 : 0;
            int i2 = (sb[3] > sb[2]) ? 3 : 2;
            float t0 = sb[i1], t1 = sb[i2];
            float mx = fmaxf(t0, t1);
            float e0 = expf(t0 - mx), e1 = expf(t1 - mx);
            float inv = 1.0f / (e0 + e1);
            ws_i[br * 2 + 0] = i1;
            ws_i[br * 2 + 1] = i2;
            ws_f[4 + br * 2 + 0] = e0 * inv;
            ws_f[4 + br * 2 + 1] = e1 * inv;
            // argmax(probs): 0 on tie (first index)
            out[DDIM + br] = (e1 > e0) ? 1.0f : 0.0f;
        }
    }
}

// ---------------------------------------------------------------------------
// Kernel 2: the bandwidth-bound part. Each block owns 16 output rows.
// 8 waves split K (512 each). For each of the 4 selected expert matrices,
// accumulate D = A(W 16x4 tile) x B(prob*x replicated over N) + C using
// V_WMMA_F32_16X16X4_F32. Single accumulator chain serves all 4 matrices
// since prob is folded into B. Cross-wave reduce via LDS; bias in epilogue.
// ---------------------------------------------------------------------------
__global__ __launch_bounds__(256) void moe_wmma_kernel(
    const float* __restrict__ rgb, const float* __restrict__ ir,
    const float* __restrict__ W_rgb, const float* __restrict__ b_rgb,
    const float* __restrict__ W_ir,  const float* __restrict__ b_ir,
    const int* __restrict__ ws_i, const float* __restrict__ ws_f,
    float* __restrict__ out)
{
    const int rowbase = blockIdx.x * 16;
    const int wave = threadIdx.x >> 5;         // 0..7
    const int lane = threadIdx.x & 31;
    const int m    = lane & 15;                // A-matrix row within tile
    const int kofs = (lane >> 4) << 1;         // lanes 0-15: K=0,1 ; 16-31: K=2,3
    const int row  = rowbase + m;
    const int k0   = wave * 512;               // this wave's K chunk (never crosses 2048)

    const float* xb = (k0 < HALF_DIM) ? (rgb + k0) : (ir + (k0 - HALF_DIM));

    v8f acc = {};
    for (int slot = 0; slot < 4; ++slot) {
        const int   e = ws_i[slot];
        const float p = ws_f[4 + slot];
        const float* W    = ((slot < 2) ? W_rgb : W_ir) + (size_t)e * DDIM * DDIM;
        const float* wrow = W + (size_t)row * DDIM + k0;
#pragma unroll 4
        for (int k = 0; k < 512; k += 4) {
            v2f a, b;
            a[0] = wrow[k + kofs];
            a[1] = wrow[k + kofs + 1];
            b[0] = p * xb[k + kofs];
            b[1] = p * xb[k + kofs + 1];
            // (neg_a, A, neg_b, B, c_mod, C, reuse_a, reuse_b)
            acc = __builtin_amdgcn_wmma_f32_16x16x4_f32(
                false, a, false, b, (short)0, acc, false, false);
        }
    }

    // C/D layout: VGPR i, lanes 0-15 -> M=i (N=lane), lanes 16-31 -> M=8+i.
    // Every column holds the same dot product; take column N=0 (lanes 0 and 16).
    __shared__ float red[8][16];
    if (lane == 0) {
#pragma unroll
        for (int i = 0; i < 8; ++i) red[wave][i] = acc[i];
    }
    if (lane == 16) {
#pragma unroll
        for (int i = 0; i < 8; ++i) red[wave][8 + i] = acc[i];
    }
    __syncthreads();

    if (threadIdx.x < 16) {
        const int r = rowbase + threadIdx.x;
        float ssum = 0.0f;
#pragma unroll
        for (int w = 0; w < 8; ++w) ssum += red[w][threadIdx.x];
#pragma unroll
        for (int slot = 0; slot < 4; ++slot) {
            const int   e = ws_i[slot];
            const float p = ws_f[4 + slot];
            const float* bias = ((slot < 2) ? b_rgb : b_ir) + e * DDIM;
            ssum = fmaf(p, bias[r], ssum);
        }
        out[r] = ssum;
    }
}

extern "C" void kernel_launch(void* const* d_in, const int* in_sizes, int n_in,
                              void* d_out, int out_size, void* d_ws, size_t ws_size,
                              hipStream_t stream) {
    const float* rgb    = (const float*)d_in[0];
    const float* ir     = (const float*)d_in[1];
    const float* Wg_rgb = (const float*)d_in[2];
    const float* bg_rgb = (const float*)d_in[3];
    const float* Wg_ir  = (const float*)d_in[4];
    const float* bg_ir  = (const float*)d_in[5];
    const float* We_rgb = (const float*)d_in[6];
    const float* be_rgb = (const float*)d_in[7];
    const float* We_ir  = (const float*)d_in[8];
    const float* be_ir  = (const float*)d_in[9];
    float* out = (float*)d_out;
    int*   ws_i = (int*)d_ws;
    float* ws_f = (float*)d_ws;

    gate_kernel<<<1, 256, 0, stream>>>(rgb, ir, Wg_rgb, bg_rgb, Wg_ir, bg_ir,
                                       out, ws_i, ws_f);
    moe_wmma_kernel<<<256, 256, 0, stream>>>(rgb, ir, We_rgb, be_rgb, We_ir, be_ir,
                                             ws_i, ws_f, out);
}